// SecondOrderSimiliarityRegulation_61117384622455
// MI455X (gfx1250) — compile-verified
//
#include <hip/hip_runtime.h>
#include <hip/hip_bf16.h>

typedef float v2f __attribute__((ext_vector_type(2)));
typedef float v8f __attribute__((ext_vector_type(8)));

#define BS   4096
#define EPSF 1e-8f

// Insert packed (value|index) into ascending top-8 list (t[0] is the min).
// For non-negative IEEE floats, unsigned compare == float compare, so the
// 12-bit row index packed into the low mantissa bits rides along for free.
__device__ __forceinline__ void top8_insert(unsigned int t[8], unsigned int pv) {
    if (pv > t[0]) {
        unsigned int c = pv;
#pragma unroll
        for (int k = 1; k < 8; ++k) {
            unsigned int lo = (c < t[k]) ? c : t[k];
            unsigned int hi = (c < t[k]) ? t[k] : c;
            t[k - 1] = lo;
            c = hi;
        }
        t[7] = c;
    }
}

__device__ __forceinline__ unsigned int pack_vi(float v, int i) {
    return (__float_as_uint(v) & 0xFFFFF000u) | (unsigned int)i;
}

// Pass 1: streaming scan. Thread = one column, blockIdx.y = row chunk.
// Produces per-(chunk,column): top-8 packed AA, top-8 packed PP, partial S_all.
__global__ void sosr_pass1(const float* __restrict__ AA, const float* __restrict__ PP,
                           unsigned int* __restrict__ wsAA, unsigned int* __restrict__ wsPP,
                           float* __restrict__ wsS, int rowsPerChunk) {
    const int j  = blockIdx.x * blockDim.x + threadIdx.x;
    const int rb = blockIdx.y;
    const int r0 = rb * rowsPerChunk;
    const int r1 = r0 + rowsPerChunk;

    unsigned int tA[8], tP[8];
#pragma unroll
    for (int k = 0; k < 8; ++k) { tA[k] = 0u; tP[k] = 0u; }
    float S = 0.0f;

    const float* pa = AA + (size_t)r0 * BS + j;
    const float* pp = PP + (size_t)r0 * BS + j;

#pragma unroll 4
    for (int i = r0; i < r1; ++i) {
        float a = __builtin_nontemporal_load(pa);   // single-use stream: TH=NT
        float p = __builtin_nontemporal_load(pp);
        pa += BS; pp += BS;
        float d = a - p + EPSF;
        S = fmaf(d, d, S);
        top8_insert(tA, pack_vi(a, i));
        top8_insert(tP, pack_vi(p, i));
    }

#pragma unroll
    for (int k = 0; k < 8; ++k) {
        wsAA[((size_t)rb * 8 + k) * BS + j] = tA[k];   // coalesced across j
        wsPP[((size_t)rb * 8 + k) * BS + j] = tP[k];
    }
    wsS[(size_t)rb * BS + j] = S;
}

// Pass 2: merge chunk partials per column, gather the <=16 AAPP values for the
// top-k union, compute SOS, then block-reduce 256 SOS values. Each wave's
// 32-value sum is done with V_WMMA_F32_16X16X4_F32 and an all-ones A matrix:
// C[m,n] = sum_k B[k,n], so sum(all C elements) == 16 * sum(B) regardless of
// fragment layout -> exact f32 reduction on the matrix pipe.
__global__ void sosr_pass2(const float* __restrict__ AA, const float* __restrict__ PP,
                           const unsigned int* __restrict__ wsAA,
                           const unsigned int* __restrict__ wsPP,
                           const float* __restrict__ wsS,
                           float* __restrict__ wsPart, int R) {
    const int j = blockIdx.x * blockDim.x + threadIdx.x;

    unsigned int tA[8], tP[8];
#pragma unroll
    for (int k = 0; k < 8; ++k) { tA[k] = 0u; tP[k] = 0u; }
    float S = 0.0f;

    for (int rb = 0; rb < R; ++rb) {
#pragma unroll
        for (int k = 0; k < 8; ++k) {
            top8_insert(tA, wsAA[((size_t)rb * 8 + k) * BS + j]);
            top8_insert(tP, wsPP[((size_t)rb * 8 + k) * BS + j]);
        }
        S += wsS[(size_t)rb * BS + j];
    }

    // Union of the two index sets; re-gather AAPP at those rows (tiny traffic).
    float U = 0.0f;
    int idxA[8];
#pragma unroll
    for (int k = 0; k < 8; ++k) {
        int m = (int)(tA[k] & 0xFFFu);
        idxA[k] = m;
        float a = AA[(size_t)m * BS + j];
        float p = PP[(size_t)m * BS + j];
        float d = a - p + EPSF;
        U = fmaf(d, d, U);
    }
#pragma unroll
    for (int k = 0; k < 8; ++k) {
        int m = (int)(tP[k] & 0xFFFu);
        bool dup = false;
#pragma unroll
        for (int q = 0; q < 8; ++q) dup = dup || (m == idxA[q]);
        if (!dup) {
            float a = AA[(size_t)m * BS + j];
            float p = PP[(size_t)m * BS + j];
            float d = a - p + EPSF;
            U = fmaf(d, d, U);
        }
    }

    // temp1 = eps*(S_all - S_union) + 1.0*S_union = eps*S_all + (1-eps)*S_union
    float temp1 = EPSF * S + (1.0f - EPSF) * U;
    float sos = sqrtf(temp1 + EPSF);

    // ---- block reduction: WMMA f32 16x16x4 with ones-A (EXEC is all 1s) ----
    v2f a2; a2[0] = 1.0f; a2[1] = 1.0f;        // all-ones 16x4 A matrix
    v2f b2; b2[0] = sos;  b2[1] = 0.0f;        // 32 live B slots + 32 zeros
    v8f c8 = {0.f, 0.f, 0.f, 0.f, 0.f, 0.f, 0.f, 0.f};
    c8 = __builtin_amdgcn_wmma_f32_16x16x4_f32(false, a2, false, b2,
                                               (short)0, c8, false, false);
    float s = 0.0f;
#pragma unroll
    for (int r = 0; r < 8; ++r) s += c8[r];
#pragma unroll
    for (int off = 16; off > 0; off >>= 1) s += __shfl_xor(s, off, 32);
    // now every lane holds 16 * (sum of this wave's 32 SOS values)

    __shared__ float wsum[8];
    const int lane = threadIdx.x & 31;
    const int wave = threadIdx.x >> 5;
    if (lane == 0) wsum[wave] = s * 0.0625f;   // /16
    __syncthreads();
    if (threadIdx.x == 0) {
        float p = 0.0f;
#pragma unroll
        for (int w = 0; w < 8; ++w) p += wsum[w];
        wsPart[blockIdx.x] = p;
    }
}

// Pass 3: fold the 16 block partials into the scalar mean.
__global__ void sosr_pass3(const float* __restrict__ wsPart, int nPart,
                           float* __restrict__ out) {
    if (threadIdx.x == 0) {
        float s = 0.0f;
        for (int i = 0; i < nPart; ++i) s += wsPart[i];
        out[0] = s * (1.0f / (float)BS);
    }
}

extern "C" void kernel_launch(void* const* d_in, const int* in_sizes, int n_in,
                              void* d_out, int out_size, void* d_ws, size_t ws_size,
                              hipStream_t stream) {
    const float* AA = (const float*)d_in[0];
    const float* PP = (const float*)d_in[1];

    // Row-split factor for occupancy: 128*R waves in pass 1. Pick the largest
    // R in {8,4,2,1} whose partials fit in d_ws.
    int R = 1;
    for (int r = 8; r >= 1; r >>= 1) {
        size_t need = ((size_t)r * 17 * BS + 64) * sizeof(unsigned int);
        if (need <= ws_size) { R = r; break; }
    }

    unsigned int* wsAA = (unsigned int*)d_ws;
    unsigned int* wsPP = wsAA + (size_t)R * 8 * BS;
    float*        wsS  = (float*)(wsPP + (size_t)R * 8 * BS);
    float*        wsPart = wsS + (size_t)R * BS;

    const int nColBlocks = BS / 256;   // 16
    dim3 g1(nColBlocks, R);
    sosr_pass1<<<g1, 256, 0, stream>>>(AA, PP, wsAA, wsPP, wsS, BS / R);
    sosr_pass2<<<nColBlocks, 256, 0, stream>>>(AA, PP, wsAA, wsPP, wsS, wsPart, R);
    sosr_pass3<<<1, 32, 0, stream>>>(wsPart, nColBlocks, (float*)d_out);
}